// STGCN_88613765251321
// MI455X (gfx1250) — compile-verified
//
#include <hip/hip_runtime.h>
#include <math.h>

// ---------------------------------------------------------------------------
// STGCN forward for MI455X (gfx1250, wave32, WMMA).
// Dominant cost: spatio graph conv as dense (R x 3072) * (3072 x 1024) GEMM,
// executed with V_WMMA_F32_16X16X4_F32 to keep fp32 precision.
// LDS staging: double-buffered GLOBAL_LOAD_ASYNC_TO_LDS_B128 (ASYNCcnt),
// next slab's fills overlap the current slab's WMMA stream.
// Workspace requirement: ~153 MB (see layout in kernel_launch).
// ---------------------------------------------------------------------------

typedef float v8f __attribute__((ext_vector_type(8)));
typedef float v2f __attribute__((ext_vector_type(2)));
typedef int   v4i_g __attribute__((vector_size(16)));   // matches builtin proto

constexpr int B_    = 8;
constexpr int T_INn = 12;
constexpr int H_OUTn= 2;
constexpr int N_    = 1024;
constexpr int KDIM  = 3 * N_;   // spatio GEMM reduction dim (3 Chebyshev terms)

#if defined(__has_builtin)
#if __has_builtin(__builtin_amdgcn_global_load_async_to_lds_b128) && \
    __has_builtin(__builtin_amdgcn_s_wait_asynccnt)
#define USE_ASYNC_LDS 1
#endif
#endif
#ifndef USE_ASYNC_LDS
#define USE_ASYNC_LDS 0
#endif

__device__ __forceinline__ float sigmoidf_(float x) {
    return 1.0f / (1.0f + expf(-x));
}

// 16B global -> LDS copy; async (no VGPR data path) when available.
__device__ __forceinline__ void lds_cp16(float* lds_dst, const float* g_src) {
#if USE_ASYNC_LDS
    __builtin_amdgcn_global_load_async_to_lds_b128(
        (__attribute__((address_space(1))) v4i_g*)(v4i_g*)(void*)g_src,
        (__attribute__((address_space(3))) v4i_g*)(v4i_g*)(void*)lds_dst,
        0, 0);
#else
    *reinterpret_cast<float4*>(lds_dst) = *reinterpret_cast<const float4*>(g_src);
#endif
}

__device__ __forceinline__ void lds_cp_drain() {
#if USE_ASYNC_LDS
    __builtin_amdgcn_s_wait_asynccnt(0);   // my wave's async fills done
#endif
}

// ---------------------------------------------------------------------------
// State packing: x_state[b,t,n,{0,1}] = {signal, time}
// ---------------------------------------------------------------------------
__global__ void pack_input_k(const float* __restrict__ in,
                             const float* __restrict__ in_time,
                             float* __restrict__ st) {
    int idx = blockIdx.x * 256 + threadIdx.x;           // over B*T*N
    if (idx >= B_ * T_INn * N_) return;
    st[(size_t)idx * 2 + 0] = in[idx];
    st[(size_t)idx * 2 + 1] = in_time[idx];
}

// h0[b,c,t,n] = st[b,t,n,c]   (B,2,T,N)
__global__ void unpack_state_k(const float* __restrict__ st,
                               float* __restrict__ h0) {
    int idx = blockIdx.x * 256 + threadIdx.x;           // over B*2*T*N
    if (idx >= B_ * 2 * T_INn * N_) return;
    int n = idx % N_;
    int t = (idx / N_) % T_INn;
    int c = (idx / (N_ * T_INn)) % 2;
    int b = idx / (N_ * T_INn * 2);
    h0[idx] = st[(((size_t)(b * T_INn + t) * N_) + n) * 2 + c];
}

// ---------------------------------------------------------------------------
// Temporal conv, fused bias + residual-align + activation.
// MODE: 0 = GLU (weights hold 2*Cout rows), 1 = relu, 2 = sigmoid
// x: (B,Cin,Tin,N)  w: (Cout{*2},Cin,kt)  out: (B,Cout,Tin-kt+1,N)
// ---------------------------------------------------------------------------
template <int MODE>
__global__ void tconv_k(const float* __restrict__ x, const float* __restrict__ w,
                        const float* __restrict__ bias, float* __restrict__ out,
                        int Cin, int Cout, int Tin, int kt) {
    const int Tout = Tin - kt + 1;
    int idx = blockIdx.x * 256 + threadIdx.x;
    int total = B_ * Cout * Tout * N_;
    if (idx >= total) return;
    int n  = idx % N_;
    int t  = (idx / N_) % Tout;
    int co = (idx / (N_ * Tout)) % Cout;
    int b  = idx / (N_ * Tout * Cout);

    const float* xb = x + (size_t)b * Cin * Tin * N_ + n;
    float acc  = bias[co];
    float acc2 = (MODE == 0) ? bias[co + Cout] : 0.0f;
    for (int ci = 0; ci < Cin; ++ci) {
        const float* xc = xb + ((size_t)ci * Tin + t) * N_;
        const float* wl = w + ((size_t)co * Cin + ci) * kt;
        const float* wh = w + ((size_t)(co + Cout) * Cin + ci) * kt;
        for (int dt = 0; dt < kt; ++dt) {
            float xv = xc[(size_t)dt * N_];
            acc = fmaf(wl[dt], xv, acc);
            if (MODE == 0) acc2 = fmaf(wh[dt], xv, acc2);
        }
    }
    // residual with channel alignment (pad when Cin < Cout), time shift kt-1
    float resid = 0.0f;
    if (co < Cin) resid = xb[((size_t)co * Tin + (t + kt - 1)) * N_];

    float r;
    if (MODE == 0)      r = (acc + resid) * sigmoidf_(acc2);
    else if (MODE == 1) r = fmaxf(acc + resid, 0.0f);
    else                r = sigmoidf_(acc + resid);
    out[idx] = r;
}

// ---------------------------------------------------------------------------
// Theta stage: y[r, k*N+m] = sum_i theta[i,o,k] * h[b,i,t,m],  r=(b*64+o)*T+t
// Produces the A matrix (R x 3072) of the spatio GEMM, row-major contiguous.
// ---------------------------------------------------------------------------
__global__ void theta_k(const float* __restrict__ h, const float* __restrict__ theta,
                        float* __restrict__ y, int Tcur) {
    int idx = blockIdx.x * 256 + threadIdx.x;           // over R*3*N
    long total = (long)B_ * 64 * Tcur * 3 * N_;
    if (idx >= total) return;
    int m = idx % N_;
    int k = (idx / N_) % 3;
    int r = idx / (3 * N_);
    int t = r % Tcur;
    int o = (r / Tcur) % 64;
    int b = r / (Tcur * 64);

    const float* hb = h + ((size_t)b * 64 * Tcur + t) * N_ + m;
    float acc = 0.0f;
    for (int i = 0; i < 64; ++i)
        acc = fmaf(theta[(i * 64 + o) * 3 + k], hb[(size_t)i * Tcur * N_], acc);
    y[(size_t)r * KDIM + (size_t)k * N_ + m] = acc;
}

// ---------------------------------------------------------------------------
// Pack lk (3,N,N) -> Bp[(k*N+m), n] = lk[k][n][m]  (LDS-tiled transpose)
// ---------------------------------------------------------------------------
__global__ void packlk_k(const float* __restrict__ lk, float* __restrict__ Bp) {
    __shared__ float tile[32][33];
    int k  = blockIdx.z;
    int n0 = blockIdx.x * 32;
    int m0 = blockIdx.y * 32;
    int tx = threadIdx.x, ty = threadIdx.y;
    tile[ty][tx] = lk[((size_t)k * N_ + (n0 + ty)) * N_ + m0 + tx];
    __syncthreads();
    Bp[((size_t)k * N_ + (m0 + ty)) * N_ + n0 + tx] = tile[tx][ty];
}

// ---------------------------------------------------------------------------
// Spatio GEMM with WMMA f32 16x16x4:
//   D[r,n] = relu( sum_q A[r,q]*Bp[q,n] + sbias[(r/T)%64] + Res[r,n] )
// 64x64 output tile per 256-thread block (8 wave32 waves: 4 in M x 2 in N,
// each wave owns one 16-row stripe x two 16-col WMMA tiles).
// K in 64-wide slabs through DOUBLE-BUFFERED LDS (2 x 34 KB of the WGP's
// 320 KB): next slab's GLOBAL_LOAD_ASYNC_TO_LDS_B128 fills are issued right
// after the barrier and overlap the current slab's 32-WMMA stream; the
// s_wait_asynccnt at the loop head covers fills issued a full compute phase
// earlier. Padded stride 68 keeps 16B alignment + conflict-free A reads.
// ---------------------------------------------------------------------------
__global__ __launch_bounds__(256) void spatio_gemm_k(
        const float* __restrict__ A, const float* __restrict__ Bp,
        const float* __restrict__ Res, const float* __restrict__ sbias,
        float* __restrict__ Out, int R, int Tcur) {
    constexpr int LDA = 68;                  // 272B row: 16B multiple, bank-skewed
    __shared__ float As[2][64 * LDA];
    __shared__ float Bs[2][64 * LDA];

    const int nBase = blockIdx.x * 64;
    const int rBase = blockIdx.y * 64;
    const int tid   = threadIdx.x;
    const int lane  = tid & 31;
    const int w     = tid >> 5;
    const int mBase = (w & 3) * 16;
    const int nB0   = (w >> 2) * 32;
    const int lane16 = lane & 15;
    const int hi     = lane >> 4;

    v8f acc0 = {0, 0, 0, 0, 0, 0, 0, 0};
    v8f acc1 = {0, 0, 0, 0, 0, 0, 0, 0};

    // Issue async fills of one 64x64 A slab + B slab into buffer `buf`.
    auto stage = [&](int buf, int k0) {
#pragma unroll
        for (int it = 0; it < 4; ++it) {
            int chunk = it * 256 + tid;
            int row   = chunk >> 4;
            int c4    = (chunk & 15) << 2;
            lds_cp16(&As[buf][row * LDA + c4],
                     A + (size_t)(rBase + row) * KDIM + k0 + c4);
            lds_cp16(&Bs[buf][row * LDA + c4],
                     Bp + (size_t)(k0 + row) * N_ + nBase + c4);
        }
    };

    stage(0, 0);                               // prologue: fill buffer 0
    for (int k0 = 0; k0 < KDIM; k0 += 64) {
        const int cur = (k0 >> 6) & 1;
        lds_cp_drain();        // my outstanding fills (current slab) complete
        __syncthreads();       // all waves' fills visible; prev slab fully read
        if (k0 + 64 < KDIM) stage(cur ^ 1, k0 + 64);   // overlap with compute

        // 16 WMMA K-substeps of 4 each. Fragment layouts per CDNA5 ISA:
        //  A 16x4: lanes 0-15 -> {K,K+1}, lanes 16-31 -> {K+2,K+3}
        //  B 4x16: same K split, N striped across lanes.
        const float* __restrict__ Ac = &As[cur][0];
        const float* __restrict__ Bc = &Bs[cur][0];
#pragma unroll
        for (int kk = 0; kk < 64; kk += 4) {
            int ks = kk + 2 * hi;
            v2f a = *reinterpret_cast<const v2f*>(&Ac[(mBase + lane16) * LDA + ks]);
            v2f b0, b1;
            b0.x = Bc[ks * LDA + nB0 + lane16];
            b0.y = Bc[(ks + 1) * LDA + nB0 + lane16];
            b1.x = Bc[ks * LDA + nB0 + 16 + lane16];
            b1.y = Bc[(ks + 1) * LDA + nB0 + 16 + lane16];
            acc0 = __builtin_amdgcn_wmma_f32_16x16x4_f32(
                false, a, false, b0, (short)0, acc0, false, false);
            acc1 = __builtin_amdgcn_wmma_f32_16x16x4_f32(
                false, a, false, b1, (short)0, acc1, false, false);
        }
    }

    // Epilogue: D layout = 8 VGPRs, lanes 0-15 -> M=v, lanes 16-31 -> M=v+8.
#pragma unroll
    for (int v = 0; v < 8; ++v) {
        int gr  = rBase + mBase + v + 8 * hi;
        if (gr >= R) continue;
        float bia = sbias[(gr / Tcur) % 64];
        int gc0 = nBase + nB0 + lane16;
        size_t i0 = (size_t)gr * N_ + gc0;
        Out[i0]      = fmaxf(acc0[v] + bia + Res[i0], 0.0f);
        Out[i0 + 16] = fmaxf(acc1[v] + bia + Res[i0 + 16], 0.0f);
    }
}

// ---------------------------------------------------------------------------
// LayerNorm over (N,C) per (b,t). x: (B,C,T,N); gamma/beta: (N,C)
// ---------------------------------------------------------------------------
__global__ void lnorm_k(const float* __restrict__ X, const float* __restrict__ g,
                        const float* __restrict__ bt, float* __restrict__ Out,
                        int C, int Tcur) {
    int b = blockIdx.x / Tcur;
    int t = blockIdx.x % Tcur;
    int tid = threadIdx.x;
    int M = C * N_;

    float s = 0.0f, s2 = 0.0f;
    for (int i = tid; i < M; i += 256) {
        int c = i >> 10, n = i & (N_ - 1);
        float v = X[((size_t)(b * C + c) * Tcur + t) * N_ + n];
        s += v; s2 += v * v;
    }
    __shared__ float sh[256], sh2[256];
    sh[tid] = s; sh2[tid] = s2;
    __syncthreads();
    for (int off = 128; off > 0; off >>= 1) {
        if (tid < off) { sh[tid] += sh[tid + off]; sh2[tid] += sh2[tid + off]; }
        __syncthreads();
    }
    float inv  = 1.0f / (float)M;
    float mean = sh[0] * inv;
    float var  = sh2[0] * inv - mean * mean;
    float rs   = rsqrtf(var + 1e-5f);
    for (int i = tid; i < M; i += 256) {
        int c = i >> 10, n = i & (N_ - 1);
        size_t xi = ((size_t)(b * C + c) * Tcur + t) * N_ + n;
        Out[xi] = (X[xi] - mean) * rs * g[n * C + c] + bt[n * C + c];
    }
}

// ---------------------------------------------------------------------------
// 1x1 output conv: out[b, step, n] = ofc_b + sum_c ofc_w[c] * H[b,c,0,n]
// ---------------------------------------------------------------------------
__global__ void head_k(const float* __restrict__ H, const float* __restrict__ w,
                       const float* __restrict__ bws, float* __restrict__ out, int s) {
    int idx = blockIdx.x * 256 + threadIdx.x;           // over B*N
    if (idx >= B_ * N_) return;
    int n = idx % N_, b = idx / N_;
    float acc = bws[0];
    for (int c = 0; c < 128; ++c)
        acc = fmaf(w[c], H[((size_t)b * 128 + c) * N_ + n], acc);
    out[((size_t)b * H_OUTn + s) * N_ + n] = acc;
}

// Shift rolling state, append prediction + ground-truth time feature.
__global__ void advance_k(const float* __restrict__ stOld, const float* __restrict__ dout,
                          const float* __restrict__ tgt_time, float* __restrict__ stNew,
                          int s) {
    int idx = blockIdx.x * 256 + threadIdx.x;           // over B*T*N
    if (idx >= B_ * T_INn * N_) return;
    int n = idx % N_;
    int t = (idx / N_) % T_INn;
    int b = idx / (N_ * T_INn);
    size_t o = (size_t)idx * 2;
    if (t < T_INn - 1) {
        size_t i2 = ((size_t)(b * T_INn + t + 1) * N_ + n) * 2;
        stNew[o]     = stOld[i2];
        stNew[o + 1] = stOld[i2 + 1];
    } else {
        stNew[o]     = dout[((size_t)b * H_OUTn + s) * N_ + n];
        stNew[o + 1] = tgt_time[((size_t)b * H_OUTn + s) * N_ + n];
    }
}

// ---------------------------------------------------------------------------
// Host orchestration
// ---------------------------------------------------------------------------
static inline void launch_tconv(int mode, const float* x, const float* w, const float* b,
                                float* out, int Cin, int Cout, int Tin, int kt,
                                hipStream_t st) {
    int Tout = Tin - kt + 1;
    int total = B_ * Cout * Tout * N_;
    int blocks = (total + 255) / 256;
    if (mode == 0)      tconv_k<0><<<blocks, 256, 0, st>>>(x, w, b, out, Cin, Cout, Tin, kt);
    else if (mode == 1) tconv_k<1><<<blocks, 256, 0, st>>>(x, w, b, out, Cin, Cout, Tin, kt);
    else                tconv_k<2><<<blocks, 256, 0, st>>>(x, w, b, out, Cin, Cout, Tin, kt);
}

extern "C" void kernel_launch(void* const* d_in, const int* in_sizes, int n_in,
                              void* d_out, int out_size, void* d_ws, size_t ws_size,
                              hipStream_t stream) {
    (void)in_sizes; (void)n_in; (void)out_size; (void)ws_size;

    const float* input       = (const float*)d_in[0];
    const float* input_time  = (const float*)d_in[2];
    const float* target_time = (const float*)d_in[3];
    const float* lk          = (const float*)d_in[4];
    const float* b1t1_w = (const float*)d_in[5];   const float* b1t1_b = (const float*)d_in[6];
    const float* b1s_th = (const float*)d_in[7];   const float* b1s_b  = (const float*)d_in[8];
    const float* b1t2_w = (const float*)d_in[9];   const float* b1t2_b = (const float*)d_in[10];
    const float* b1ln_g = (const float*)d_in[11];  const float* b1ln_b = (const float*)d_in[12];
    const float* b2t1_w = (const float*)d_in[13];  const float* b2t1_b = (const float*)d_in[14];
    const float* b2s_th = (const float*)d_in[15];  const float* b2s_b  = (const float*)d_in[16];
    const float* b2t2_w = (const float*)d_in[17];  const float* b2t2_b = (const float*)d_in[18];
    const float* b2ln_g = (const float*)d_in[19];  const float* b2ln_b = (const float*)d_in[20];
    const float* ot1_w  = (const float*)d_in[21];  const float* ot1_b  = (const float*)d_in[22];
    const float* oln_g  = (const float*)d_in[23];  const float* oln_b  = (const float*)d_in[24];
    const float* ot2_w  = (const float*)d_in[25];  const float* ot2_b  = (const float*)d_in[26];
    const float* ofc_w  = (const float*)d_in[27];  const float* ofc_b  = (const float*)d_in[28];

    // Workspace carve (floats). Total ~38.1M floats (~153 MB).
    float* ws   = (float*)d_ws;
    float* stA  = ws;                       // (B,12,N,2)       196608
    float* stB  = stA  + 196608;            //                  196608
    float* Bp   = stB  + 196608;            // packed lk        3145728
    float* bufA = Bp   + 3145728;           // scratch          6291456
    float* bufB = bufA + 6291456;           // scratch          6291456
    float* bufC = bufB + 6291456;           // scratch          6291456
    float* bufY = bufC + 6291456;           // GEMM A matrix    15728640

    float* out = (float*)d_out;

    pack_input_k<<<(B_ * T_INn * N_) / 256, 256, 0, stream>>>(input, input_time, stA);
    packlk_k<<<dim3(N_ / 32, N_ / 32, 3), dim3(32, 32), 0, stream>>>(lk, Bp);

    for (int s = 0; s < H_OUTn; ++s) {
        float* stIn  = (s == 0) ? stA : stB;
        float* stOut = (s == 0) ? stB : stA;

        // (B,T,N,2) -> (B,2,T,N)
        unpack_state_k<<<(B_ * 2 * T_INn * N_) / 256, 256, 0, stream>>>(stIn, bufA);

        // ---- ST block 1 ----
        launch_tconv(0, bufA, b1t1_w, b1t1_b, bufB, 2, 64, 12, 3, stream);   // GLU -> (B,64,10,N)
        {
            int T1 = 10, R1 = B_ * 64 * T1;
            theta_k<<<((long)R1 * 3 * N_) / 256, 256, 0, stream>>>(bufB, b1s_th, bufY, T1);
            spatio_gemm_k<<<dim3(N_ / 64, R1 / 64), 256, 0, stream>>>(
                bufY, Bp, bufB, b1s_b, bufC, R1, T1);                  // relu(+res) -> (B,64,10,N)
        }
        launch_tconv(1, bufC, b1t2_w, b1t2_b, bufA, 64, 64, 10, 3, stream);  // relu -> (B,64,8,N)
        lnorm_k<<<B_ * 8, 256, 0, stream>>>(bufA, b1ln_g, b1ln_b, bufB, 64, 8);

        // ---- ST block 2 ----
        launch_tconv(0, bufB, b2t1_w, b2t1_b, bufC, 64, 64, 8, 3, stream);   // GLU -> (B,64,6,N)
        {
            int T2 = 6, R2 = B_ * 64 * T2;
            theta_k<<<((long)R2 * 3 * N_) / 256, 256, 0, stream>>>(bufC, b2s_th, bufY, T2);
            spatio_gemm_k<<<dim3(N_ / 64, R2 / 64), 256, 0, stream>>>(
                bufY, Bp, bufC, b2s_b, bufA, R2, T2);                  // -> (B,64,6,N)
        }
        launch_tconv(1, bufA, b2t2_w, b2t2_b, bufB, 64, 128, 6, 3, stream);  // relu -> (B,128,4,N)
        lnorm_k<<<B_ * 4, 256, 0, stream>>>(bufB, b2ln_g, b2ln_b, bufC, 128, 4);

        // ---- Output layer ----
        launch_tconv(0, bufC, ot1_w, ot1_b, bufA, 128, 128, 4, 4, stream);   // GLU -> (B,128,1,N)
        lnorm_k<<<B_ * 1, 256, 0, stream>>>(bufA, oln_g, oln_b, bufB, 128, 1);
        launch_tconv(2, bufB, ot2_w, ot2_b, bufC, 128, 128, 1, 1, stream);   // sigmoid -> (B,128,1,N)
        head_k<<<(B_ * N_) / 256, 256, 0, stream>>>(bufC, ofc_w, ofc_b, out, s);

        // roll state forward
        advance_k<<<(B_ * T_INn * N_) / 256, 256, 0, stream>>>(stIn, out, target_time, stOut, s);
    }
}